// DynamicHyperCell_36833639531045
// MI455X (gfx1250) — compile-verified
//
#include <hip/hip_runtime.h>

// Dims (fixed by the reference)
#define B_    64
#define T_    512
#define IN_   64
#define H_    512
#define HYP_  256
#define NZ_   64
#define GH_   768      // 3*HYP
#define XHAT_ 576      // H+IN

#define NWG    64
#define NTHR   256
#define NWAVES (NWG * (NTHR / 32))   // 512 waves

typedef __attribute__((ext_vector_type(2))) float v2f;
typedef __attribute__((ext_vector_type(8))) float v8f;

__device__ __forceinline__ float sigmf(float x) { return 1.0f / (1.0f + expf(-x)); }

// ---------------------------------------------------------------------------
// f32 WMMA 16x16 tile: C += A_tile(16xK) * W_tile(16xK)^T, K stepped by 4.
// A layout (ISA 7.12.2, 32-bit A 16x4): lane holds row M=lane%16, VGPR v holds
// K = k0 + v + 2*(lane>=16).  B (4x16) is symmetric: lane holds col N=lane%16,
// VGPR v holds K = k0 + v + 2*(lane>=16).  Both fragments therefore load as a
// single per-lane float2 from row-major [rows, K] storage.
// ---------------------------------------------------------------------------
__device__ __forceinline__ v8f wmma_acc(v8f c, const float* __restrict__ A, int lda,
                                        const float* __restrict__ W, int ldw,
                                        int K, int m0, int n0) {
  const int lane  = threadIdx.x & 31;
  const int half2 = (lane >> 4) << 1;     // 0 or 2
  const int r     = lane & 15;
  const float* ap = A + (size_t)(m0 + r) * lda + half2;
  const float* wp = W + (size_t)(n0 + r) * ldw + half2;
  for (int k = 0; k < K; k += 4) {
    v2f av = *(const v2f*)(ap + k);
    v2f bv = *(const v2f*)(wp + k);
    c = __builtin_amdgcn_wmma_f32_16x16x4_f32(false, av, false, bv, (short)0, c,
                                              false, false);
  }
  return c;
}

// C/D layout: VGPR v, lanes 0-15 -> M=v, lanes 16-31 -> M=v+8; N = lane%16.
__device__ __forceinline__ void store_tile(float* __restrict__ C, int ldc, int m0,
                                           int n0, v8f c,
                                           const float* __restrict__ bias) {
  const int lane   = threadIdx.x & 31;
  const int rowoff = (lane >> 4) * 8;
  const int col    = n0 + (lane & 15);
  float bv = bias ? bias[col] : 0.0f;
#pragma unroll
  for (int v = 0; v < 8; ++v)
    C[(size_t)(m0 + rowoff + v) * ldc + col] = c[v] + bv;
}

// 4-way block reduction (sum) over 256 threads via LDS.
__device__ __forceinline__ void block_reduce4(float& a, float& b, float& c, float& d) {
  __shared__ float s0[NTHR], s1[NTHR], s2[NTHR], s3[NTHR];
  const int tid = threadIdx.x;
  s0[tid] = a; s1[tid] = b; s2[tid] = c; s3[tid] = d;
  __syncthreads();
  for (int off = NTHR / 2; off > 0; off >>= 1) {
    if (tid < off) {
      s0[tid] += s0[tid + off]; s1[tid] += s1[tid + off];
      s2[tid] += s2[tid + off]; s3[tid] += s3[tid + off];
    }
    __syncthreads();
  }
  a = s0[0]; b = s1[0]; c = s2[0]; d = s3[0];
  __syncthreads();
}

// Grid-wide sense-reversal barrier (all NWG workgroups are co-resident).
__device__ __forceinline__ void grid_barrier(unsigned* bar) {
  __syncthreads();
  if (threadIdx.x == 0) {
    __threadfence();
    unsigned gen = __hip_atomic_load(&bar[1], __ATOMIC_RELAXED, __HIP_MEMORY_SCOPE_AGENT);
    unsigned prev = __hip_atomic_fetch_add(&bar[0], 1u, __ATOMIC_ACQ_REL, __HIP_MEMORY_SCOPE_AGENT);
    if (prev == NWG - 1) {
      __hip_atomic_store(&bar[0], 0u, __ATOMIC_RELAXED, __HIP_MEMORY_SCOPE_AGENT);
      __hip_atomic_fetch_add(&bar[1], 1u, __ATOMIC_ACQ_REL, __HIP_MEMORY_SCOPE_AGENT);
    } else {
      while (__hip_atomic_load(&bar[1], __ATOMIC_RELAXED, __HIP_MEMORY_SCOPE_AGENT) == gen)
        __builtin_amdgcn_s_sleep(2);
    }
    __threadfence();
  }
  __syncthreads();
}

// ---------------------------------------------------------------------------
// Setup kernel (runs once per launch): fuse z->d projections into A_* / b_*,
//   A_m[g][n][k] = sum_z d*_w[g,n,z] * z*_w[g*64+z,k]
//   b_h[g][n]    = sum_z zh_b[g*64+z] * dh_w[g,n,z]  (likewise b_x; b_b = db_b)
// and initialize h, hh, barrier.
// ---------------------------------------------------------------------------
__global__ void hyper_setup(const float* __restrict__ h0, const float* __restrict__ hhat0,
                            const float* __restrict__ zh_w, const float* __restrict__ zh_b,
                            const float* __restrict__ zx_w, const float* __restrict__ zx_b,
                            const float* __restrict__ zb_w, const float* __restrict__ dh_w,
                            const float* __restrict__ dx_w, const float* __restrict__ db_w,
                            const float* __restrict__ db_b,
                            float* __restrict__ h, float* __restrict__ hh,
                            float* __restrict__ Ah, float* __restrict__ Ax,
                            float* __restrict__ Ab, float* __restrict__ bh,
                            float* __restrict__ bx, float* __restrict__ bb,
                            unsigned* __restrict__ bar) {
  long i = (long)blockIdx.x * NTHR + threadIdx.x;
  const long NA = 3L * 3 * H_ * HYP_;               // 1,179,648
  if (i < NA) {
    int mat = (int)(i / (3L * H_ * HYP_));
    long r  = i % (3L * H_ * HYP_);
    int g   = (int)(r / ((long)H_ * HYP_));
    long r2 = r % ((long)H_ * HYP_);
    int n = (int)(r2 >> 8), k = (int)(r2 & 255);
    const float* dW = mat == 0 ? dh_w : (mat == 1 ? dx_w : db_w);
    const float* zW = mat == 0 ? zh_w : (mat == 1 ? zx_w : zb_w);
    float* out      = mat == 0 ? Ah   : (mat == 1 ? Ax   : Ab);
    const float* dp = dW + ((size_t)g * H_ + n) * NZ_;
    const float* zp = zW + (size_t)g * NZ_ * HYP_ + k;
    float acc = 0.0f;
    for (int z = 0; z < NZ_; ++z) acc += dp[z] * zp[(size_t)z * HYP_];
    out[((size_t)g * H_ + n) * HYP_ + k] = acc;
    return;
  }
  i -= NA;
  if (i < 3 * H_) {  // bias_h
    int g = (int)(i / H_), n = (int)(i % H_);
    const float* dp = dh_w + ((size_t)g * H_ + n) * NZ_;
    float acc = 0.0f;
    for (int z = 0; z < NZ_; ++z) acc += zh_b[g * NZ_ + z] * dp[z];
    bh[i] = acc; return;
  }
  i -= 3 * H_;
  if (i < 3 * H_) {  // bias_x
    int g = (int)(i / H_), n = (int)(i % H_);
    const float* dp = dx_w + ((size_t)g * H_ + n) * NZ_;
    float acc = 0.0f;
    for (int z = 0; z < NZ_; ++z) acc += zx_b[g * NZ_ + z] * dp[z];
    bx[i] = acc; return;
  }
  i -= 3 * H_;
  if (i < 3 * H_) { bb[i] = db_b[i]; return; }      // zb has no bias -> only db_b
  i -= 3 * H_;
  if (i < B_ * H_)  { h[i]  = h0[i];    return; }
  i -= B_ * H_;
  if (i < B_ * HYP_){ hh[i] = hhat0[i]; return; }
  i -= B_ * HYP_;
  if (i < 2) { bar[i] = 0u; return; }
}

// ---------------------------------------------------------------------------
// Persistent kernel: entire T=512 recurrence, 4 grid barriers per step.
// ---------------------------------------------------------------------------
__global__ void __launch_bounds__(NTHR, 1)
hyper_persistent(const float* __restrict__ xs, const float* __restrict__ cond,
                 const float* __restrict__ hx2h_w, const float* __restrict__ hx2h_b,
                 const float* __restrict__ hh2h_w, const float* __restrict__ hh2h_b,
                 const float* __restrict__ wh, const float* __restrict__ wx,
                 const float* __restrict__ ln_g, const float* __restrict__ ln_b,
                 const float* __restrict__ Ah, const float* __restrict__ Ax,
                 const float* __restrict__ Ab, const float* __restrict__ bh,
                 const float* __restrict__ bx, const float* __restrict__ bb,
                 float* __restrict__ h, float* __restrict__ hh,
                 float* __restrict__ ihp, float* __restrict__ hpp,
                 float* __restrict__ dH, float* __restrict__ dX, float* __restrict__ dB,
                 float* __restrict__ phl, float* __restrict__ pxl,
                 unsigned* __restrict__ bar, float* __restrict__ ys) {
  const int tid   = threadIdx.x;
  const int wg    = blockIdx.x;                  // 0..63
  const int gwave = wg * (NTHR / 32) + (tid >> 5);  // 0..511

  for (int t = 0; t < T_; ++t) {
    // ---- P1: all GEMMs depending on (h, hh, x_t): 1152 16x16 tiles --------
    for (int tile = gwave; tile < 1152; tile += NWAVES) {
      v8f c = {0.f, 0.f, 0.f, 0.f, 0.f, 0.f, 0.f, 0.f};
      if (tile < 192) {                          // ih = [h,cond] @ hx2h_w^T + b
        int m0 = (tile & 3) * 16, n0 = (tile >> 2) * 16;
        c = wmma_acc(c, h, H_, hx2h_w, XHAT_, H_, m0, n0);
        c = wmma_acc(c, cond, IN_, hx2h_w + H_, XHAT_, IN_, m0, n0);
        store_tile(ihp, GH_, m0, n0, c, hx2h_b);
      } else if (tile < 384) {                   // hhp = hh @ hh2h_w^T + b
        int l = tile - 192;
        int m0 = (l & 3) * 16, n0 = (l >> 2) * 16;
        c = wmma_acc(c, hh, HYP_, hh2h_w, HYP_, HYP_, m0, n0);
        store_tile(hpp, GH_, m0, n0, c, hh2h_b);
      } else if (tile < 768) {                   // ph_lin[g] = h @ wh[g]^T
        int l = tile - 384; int g = l >> 7; int l2 = l & 127;
        int m0 = (l2 & 3) * 16, n0 = (l2 >> 2) * 16;
        c = wmma_acc(c, h, H_, wh + (size_t)g * H_ * H_, H_, H_, m0, n0);
        store_tile(phl + g * H_, 3 * H_, m0, n0, c, nullptr);
      } else {                                   // px_lin[g] = x_t @ wx[g]^T
        int l = tile - 768; int g = l >> 7; int l2 = l & 127;
        int m0 = (l2 & 3) * 16, n0 = (l2 >> 2) * 16;
        c = wmma_acc(c, xs + (size_t)t * IN_, T_ * IN_,
                     wx + (size_t)g * H_ * IN_, IN_, IN_, m0, n0);
        store_tile(pxl + g * H_, 3 * H_, m0, n0, c, nullptr);
      }
    }
    grid_barrier(bar);

    // ---- P2: hyper GRU gates (LN no-affine), hh updated in place ----------
    {
      const int b = wg;
      const float* ir = ihp + (size_t)b * GH_;
      const float* hr = hpp + (size_t)b * GH_;
      float a0 = ir[tid], a1 = ir[tid + 256], a2 = ir[tid + 512];
      float c0 = hr[tid], c1 = hr[tid + 256], c2 = hr[tid + 512];
      float sI = a0 + a1 + a2, qI = a0 * a0 + a1 * a1 + a2 * a2;
      float sH = c0 + c1 + c2, qH = c0 * c0 + c1 * c1 + c2 * c2;
      block_reduce4(sI, qI, sH, qH);
      float mI = sI * (1.0f / GH_), vI = qI * (1.0f / GH_) - mI * mI;
      float mH = sH * (1.0f / GH_), vH = qH * (1.0f / GH_) - mH * mH;
      float iI = rsqrtf(vI + 1e-5f), iH = rsqrtf(vH + 1e-5f);
      float i_r = (a0 - mI) * iI, i_i = (a1 - mI) * iI, i_n = (a2 - mI) * iI;
      float h_r = (c0 - mH) * iH, h_i = (c1 - mH) * iH, h_n = (c2 - mH) * iH;
      float rg = sigmf(i_r + h_r);
      float ig = sigmf(i_i + h_i);
      float ng = tanhf(i_n + rg * h_n);
      float hv = hh[(size_t)b * HYP_ + tid];
      hh[(size_t)b * HYP_ + tid] = ng + ig * (hv - ng);
    }
    grid_barrier(bar);

    // ---- P3: fused d projections: d_* = hh @ A_*^T + b_* (1152 tiles) -----
    for (int tile = gwave; tile < 1152; tile += NWAVES) {
      int mat = tile / 384, l = tile % 384;
      int g = l >> 7, l2 = l & 127;
      int m0 = (l2 & 3) * 16, n0 = (l2 >> 2) * 16;
      const float* Aw = (mat == 0 ? Ah : (mat == 1 ? Ax : Ab)) + (size_t)g * H_ * HYP_;
      const float* bw = (mat == 0 ? bh : (mat == 1 ? bx : bb)) + g * H_;
      float* Cw = (mat == 0 ? dH : (mat == 1 ? dX : dB)) + g * H_;
      v8f c = {0.f, 0.f, 0.f, 0.f, 0.f, 0.f, 0.f, 0.f};
      c = wmma_acc(c, hh, HYP_, Aw, HYP_, HYP_, m0, n0);
      store_tile(Cw, 3 * H_, m0, n0, c, bw);
    }
    grid_barrier(bar);

    // ---- P4: main cell gates with affine LNs (r feeds n), h + output ------
    {
      const int b = wg;
      const float* dHr = dH + (size_t)b * 3 * H_;
      const float* dXr = dX + (size_t)b * 3 * H_;
      const float* dBr = dB + (size_t)b * 3 * H_;
      const float* pH  = phl + (size_t)b * 3 * H_;
      const float* pX  = pxl + (size_t)b * 3 * H_;
      const int j0 = tid, j1 = tid + 256;
      float sr0 = sigmf(dHr[j0] * pH[j0] + dXr[j0] * pX[j0] + dBr[j0]);
      float sr1 = sigmf(dHr[j1] * pH[j1] + dXr[j1] * pX[j1] + dBr[j1]);
      float si0 = sigmf(dHr[512 + j0] * pH[512 + j0] + dXr[512 + j0] * pX[512 + j0] + dBr[512 + j0]);
      float si1 = sigmf(dHr[512 + j1] * pH[512 + j1] + dXr[512 + j1] * pX[512 + j1] + dBr[512 + j1]);
      float Sr = sr0 + sr1, Qr = sr0 * sr0 + sr1 * sr1;
      float Si = si0 + si1, Qi = si0 * si0 + si1 * si1;
      block_reduce4(Sr, Qr, Si, Qi);
      float mr = Sr * (1.0f / H_), vr = Qr * (1.0f / H_) - mr * mr, ivr = rsqrtf(vr + 1e-5f);
      float mi = Si * (1.0f / H_), vi = Qi * (1.0f / H_) - mi * mi, ivi = rsqrtf(vi + 1e-5f);
      float r0 = (sr0 - mr) * ivr * ln_g[j0] + ln_b[j0];
      float r1 = (sr1 - mr) * ivr * ln_g[j1] + ln_b[j1];
      float i0 = (si0 - mi) * ivi * ln_g[512 + j0] + ln_b[512 + j0];
      float i1 = (si1 - mi) * ivi * ln_g[512 + j1] + ln_b[512 + j1];
      float tn0 = tanhf(r0 * (dHr[1024 + j0] * pH[1024 + j0]) +
                        dXr[1024 + j0] * pX[1024 + j0] + dBr[1024 + j0]);
      float tn1 = tanhf(r1 * (dHr[1024 + j1] * pH[1024 + j1]) +
                        dXr[1024 + j1] * pX[1024 + j1] + dBr[1024 + j1]);
      float Sn = tn0 + tn1, Qn = tn0 * tn0 + tn1 * tn1, z2 = 0.f, z3 = 0.f;
      block_reduce4(Sn, Qn, z2, z3);
      float mn = Sn * (1.0f / H_), vn = Qn * (1.0f / H_) - mn * mn, ivn = rsqrtf(vn + 1e-5f);
      float n0v = (tn0 - mn) * ivn * ln_g[1024 + j0] + ln_b[1024 + j0];
      float n1v = (tn1 - mn) * ivn * ln_g[1024 + j1] + ln_b[1024 + j1];
      float h0v = h[(size_t)b * H_ + j0], h1v = h[(size_t)b * H_ + j1];
      float hn0 = n0v + i0 * (h0v - n0v);
      float hn1 = n1v + i1 * (h1v - n1v);
      h[(size_t)b * H_ + j0] = hn0;
      h[(size_t)b * H_ + j1] = hn1;
      ys[((size_t)b * T_ + t) * H_ + j0] = hn0;
      ys[((size_t)b * T_ + t) * H_ + j1] = hn1;
    }
    grid_barrier(bar);
  }
}

extern "C" void kernel_launch(void* const* d_in, const int* in_sizes, int n_in,
                              void* d_out, int out_size, void* d_ws, size_t ws_size,
                              hipStream_t stream) {
  const float* xs     = (const float*)d_in[0];
  const float* cond   = (const float*)d_in[1];
  const float* h0     = (const float*)d_in[2];
  const float* hhat0  = (const float*)d_in[3];
  const float* hx2h_w = (const float*)d_in[4];
  const float* hx2h_b = (const float*)d_in[5];
  const float* hh2h_w = (const float*)d_in[6];
  const float* hh2h_b = (const float*)d_in[7];
  const float* zh_w   = (const float*)d_in[8];
  const float* zh_b   = (const float*)d_in[9];
  const float* zx_w   = (const float*)d_in[10];
  const float* zx_b   = (const float*)d_in[11];
  const float* zb_w   = (const float*)d_in[12];
  const float* dh_w   = (const float*)d_in[13];
  const float* dx_w   = (const float*)d_in[14];
  const float* db_w   = (const float*)d_in[15];
  const float* db_b   = (const float*)d_in[16];
  const float* wh     = (const float*)d_in[17];
  const float* wx     = (const float*)d_in[18];
  const float* ln_g   = (const float*)d_in[19];
  const float* ln_b   = (const float*)d_in[20];
  float* ys = (float*)d_out;

  // Workspace layout (floats); ~7.3 MB total.
  float* base   = (float*)d_ws;
  unsigned* bar = (unsigned*)d_ws;               // bar[0]=cnt, bar[1]=gen
  float* h   = base + 16;
  float* hh  = h   + B_ * H_;                    // 32768
  float* ihp = hh  + B_ * HYP_;                  // 16384
  float* hpp = ihp + B_ * GH_;                   // 49152
  float* dH  = hpp + B_ * GH_;
  float* dX  = dH  + B_ * 3 * H_;                // 98304 each
  float* dB  = dX  + B_ * 3 * H_;
  float* phl = dB  + B_ * 3 * H_;
  float* pxl = phl + B_ * 3 * H_;
  float* Ah  = pxl + B_ * 3 * H_;
  float* Ax  = Ah  + 3 * H_ * HYP_;              // 393216 each
  float* Ab  = Ax  + 3 * H_ * HYP_;
  float* bh  = Ab  + 3 * H_ * HYP_;
  float* bx  = bh  + 3 * H_;
  float* bb  = bx  + 3 * H_;

  // One-time (per launch) weight fusion + state init.
  const long total = 3L * 3 * H_ * HYP_ + 3L * 3 * H_ + B_ * H_ + B_ * HYP_ + 2;
  const int blocks = (int)((total + NTHR - 1) / NTHR);
  hyper_setup<<<blocks, NTHR, 0, stream>>>(h0, hhat0, zh_w, zh_b, zx_w, zx_b, zb_w,
                                           dh_w, dx_w, db_w, db_b,
                                           h, hh, Ah, Ax, Ab, bh, bx, bb, bar);

  // Entire 512-step recurrence in one persistent cooperative kernel.
  hyper_persistent<<<NWG, NTHR, 0, stream>>>(xs, cond, hx2h_w, hx2h_b, hh2h_w, hh2h_b,
                                             wh, wx, ln_g, ln_b, Ah, Ax, Ab, bh, bx, bb,
                                             h, hh, ihp, hpp, dH, dX, dB, phl, pxl,
                                             bar, ys);
}